// _ProposalLayer_1889785610971
// MI455X (gfx1250) — compile-verified
//
#include <hip/hip_runtime.h>
#include <stdint.h>

#define K_PRE  6000
#define K_POST 300
#define NMS_TH 0.7f
#define TILES  375      // K_PRE / 16
#define WORDS  188      // ceil(K_PRE / 32)
#define COLS16 376      // u16 columns per mask row (padded even)

typedef unsigned int       u32;
typedef unsigned long long u64;
typedef unsigned short     u16;

typedef __attribute__((ext_vector_type(2))) float v2f;
typedef __attribute__((ext_vector_type(8))) float v8f;

__device__ __forceinline__ u32 fkey(float f) {
  u32 b = __float_as_uint(f);
  u32 m = (b & 0x80000000u) ? 0xFFFFFFFFu : 0x80000000u;
  return b ^ m;   // monotone: larger float -> larger key
}

// ---------------- init: zero hist, counters; state = {prefix=0, remain=K} ----
__global__ void k_init(u32* hist, u32* state, u32* cnt, int B) {
  int t = threadIdx.x;
  for (int i = t; i < B * 256; i += blockDim.x) hist[i] = 0;
  if (t < B) { state[2*t] = 0u; state[2*t+1] = K_PRE; cnt[t] = 0u; }
}

// ---------------- sortable keys from scores[:, :, 1] -------------------------
__global__ void k_keys(const float* __restrict__ scores, u32* __restrict__ ukeys,
                       int total) {
  int idx = blockIdx.x * blockDim.x + threadIdx.x;
  if (idx >= total) return;
  ukeys[idx] = fkey(scores[(size_t)idx * 2 + 1]);
}

// ---------------- MSB radix-select: histogram one byte level -----------------
__global__ void k_hist(const u32* __restrict__ ukeys, u32* __restrict__ hist,
                       const u32* __restrict__ state, int level, int N) {
  int b = blockIdx.y;
  int i = blockIdx.x * blockDim.x + threadIdx.x;
  if (i >= N) return;
  u32 key = ukeys[(size_t)b * N + i];
  if (level > 0 && (key >> (32 - 8 * level)) != state[2*b]) return;
  int bin = (key >> (24 - 8 * level)) & 255;
  atomicAdd(&hist[b * 256 + bin], 1u);
}

// ---------------- pick bin containing the K-th largest; zero hist ------------
__global__ void k_find(u32* hist, u32* state) {
  int b = blockIdx.x, t = threadIdx.x;
  __shared__ u32 h[256];
  h[t] = hist[b * 256 + t];
  hist[b * 256 + t] = 0;      // ready for next level
  __syncthreads();
  if (t == 0) {
    u32 remain = state[2*b + 1];
    u32 cum = 0; int bin = 0;
    for (int x = 255; x >= 0; --x) {
      u32 c = h[x];
      if (cum + c >= remain) { bin = x; break; }
      cum += c;
    }
    state[2*b]     = (state[2*b] << 8) | (u32)bin;
    state[2*b + 1] = remain - cum;
  }
}

// ---------------- compact all keys >= threshold into (key<<32 | ~idx) --------
__global__ void k_compact(const u32* __restrict__ ukeys, const u32* __restrict__ state,
                          u32* cnt, u64* __restrict__ comp, int N) {
  int b = blockIdx.y;
  int i = blockIdx.x * blockDim.x + threadIdx.x;
  if (i >= N) return;
  u32 key = ukeys[(size_t)b * N + i];
  if (key >= state[2*b]) {
    u32 pos = atomicAdd(&cnt[b], 1u);
    if (pos < 8192) comp[(size_t)b * 8192 + pos] = ((u64)key << 32) | (u32)(~(u32)i);
  }
}

// ---------------- one-block bitonic sort of 8192 u64, descending -------------
__global__ __launch_bounds__(1024) void k_sort(const u64* __restrict__ comp,
                                               const u32* __restrict__ cnt,
                                               u32* __restrict__ order) {
  __shared__ u64 sh[8192];   // 64 KB LDS (CDNA5 WGP has 320 KB)
  int b = blockIdx.x, t = threadIdx.x;
  u32 c = cnt[b]; if (c > 8192u) c = 8192u;
  for (int k = 0; k < 8; ++k) {
    int g = t + k * 1024;
    sh[g] = (g < (int)c) ? comp[(size_t)b * 8192 + g] : 0ull;
  }
  __syncthreads();
  for (int size = 2; size <= 8192; size <<= 1) {
    for (int stride = size >> 1; stride > 0; stride >>= 1) {
      for (int k = 0; k < 4; ++k) {
        int p = t + k * 1024;
        int l = 2 * p - (p & (stride - 1));
        bool desc = (l & size) == 0;
        u64 a = sh[l], d = sh[l + stride];
        if (desc ? (a < d) : (a > d)) { sh[l] = d; sh[l + stride] = a; }
      }
      __syncthreads();
    }
  }
  for (int k = 0; k < 8; ++k) {
    int r = t + k * 1024;
    if (r < K_PRE) order[(size_t)b * K_PRE + r] = ~((u32)(sh[r] & 0xFFFFFFFFull));
  }
}

// ---------------- gather selected anchors, decode + clip both box sets -------
__global__ void k_gather(const u32* __restrict__ order, const float* __restrict__ anchors,
                         const float* __restrict__ deltas, const float* __restrict__ im_info,
                         float* __restrict__ boxesL, float* __restrict__ areasL,
                         float* __restrict__ boxesR, float* __restrict__ areasR,
                         int N, int total) {
  int idx = blockIdx.x * blockDim.x + threadIdx.x;
  if (idx >= total) return;
  int b = idx / K_PRE;
  u32 a = order[idx];
  float x1 = anchors[(size_t)a*4+0], y1 = anchors[(size_t)a*4+1];
  float x2 = anchors[(size_t)a*4+2], y2 = anchors[(size_t)a*4+3];
  float w  = x2 - x1 + 1.0f, h = y2 - y1 + 1.0f;
  float cx = x1 + 0.5f * w,  cy = y1 + 0.5f * h;
  const float* d = deltas + ((size_t)b * N + a) * 6;
  float Hm1 = im_info[b*3+0] - 1.0f, Wm1 = im_info[b*3+1] - 1.0f;
  float pcy = d[1] * h + cy;
  float ph  = expf(d[3]) * h;
  float by1 = fminf(fmaxf(pcy - 0.5f * ph, 0.0f), Hm1);
  float by2 = fminf(fmaxf(pcy + 0.5f * ph, 0.0f), Hm1);
  { // left: dx=d0, dw=d2
    float pcx = d[0] * w + cx, pw = expf(d[2]) * w;
    float bx1 = fminf(fmaxf(pcx - 0.5f * pw, 0.0f), Wm1);
    float bx2 = fminf(fmaxf(pcx + 0.5f * pw, 0.0f), Wm1);
    float* o = boxesL + (size_t)idx * 4;
    o[0]=bx1; o[1]=by1; o[2]=bx2; o[3]=by2;
    areasL[idx] = (bx2 - bx1 + 1.0f) * (by2 - by1 + 1.0f);
  }
  { // right: dx=d4, dw=d5
    float pcx = d[4] * w + cx, pw = expf(d[5]) * w;
    float bx1 = fminf(fmaxf(pcx - 0.5f * pw, 0.0f), Wm1);
    float bx2 = fminf(fmaxf(pcx + 0.5f * pw, 0.0f), Wm1);
    float* o = boxesR + (size_t)idx * 4;
    o[0]=bx1; o[1]=by1; o[2]=bx2; o[3]=by2;
    areasR[idx] = (bx2 - bx1 + 1.0f) * (by2 - by1 + 1.0f);
  }
}

// ---------------- 16x16 IoU-mask tiles: async->LDS staging + WMMA area-sum ---
__global__ __launch_bounds__(32) void k_mask(const float* __restrict__ boxesL,
                                             const float* __restrict__ areasL,
                                             const float* __restrict__ boxesR,
                                             const float* __restrict__ areasR,
                                             u16* __restrict__ maskBase) {
  int jt = blockIdx.x, it = blockIdx.y, z = blockIdx.z;
  if (jt < it) return;                       // upper triangle only (uniform exit)
  int b = z >> 1, side = z & 1;
  const float* boxes = (side ? boxesR : boxesL) + (size_t)b * K_PRE * 4;
  const float* areas = (side ? areasR : areasL) + (size_t)b * K_PRE;
  u16* mask16 = maskBase + (size_t)z * K_PRE * COLS16;
  int i0 = it * 16, j0 = jt * 16;
  int lane = threadIdx.x;

  // LDS: [0,64)=row boxes, [64,128)=col boxes, [128,144)=row areas, [144,160)=col areas
  __shared__ float sh[160];
  #pragma unroll
  for (int s = 0; s < 5; ++s) {
    int w = lane + 32 * s;
    const float* src;
    if (w < 64)       src = boxes + (i0 * 4 + w);
    else if (w < 128) src = boxes + (j0 * 4 + (w - 64));
    else if (w < 144) src = areas + (i0 + (w - 128));
    else              src = areas + (j0 + (w - 144));
    u32 ldsAddr = (u32)(uintptr_t)(&sh[w]);  // low 32 bits of flat shared ptr = LDS offset
    asm volatile("global_load_async_to_lds_b32 %0, %1, off"
                 :: "v"(ldsAddr), "v"((u64)(uintptr_t)src) : "memory");
  }
  asm volatile("s_wait_asynccnt 0" ::: "memory");

  int n = lane & 15;
  int halfOff = (lane >> 4) << 3;            // 0 or 8 (C/D layout: M = r + 8*(lane>=16))
  float cx1 = sh[64 + n*4+0], cy1 = sh[64 + n*4+1];
  float cx2 = sh[64 + n*4+2], cy2 = sh[64 + n*4+3];
  float colA = sh[144 + n];

  // D[m][n] = area_row[m] + area_col[n]: A = ones (K-permutation immune),
  // B holds area_col in exactly one 16-lane row slot, C broadcasts area_row.
  v2f A = {1.0f, 1.0f};
  v2f Bm = { (lane < 16) ? colA : 0.0f, 0.0f };
  v8f C;
  #pragma unroll
  for (int r = 0; r < 8; ++r) C[r] = sh[128 + halfOff + r];
  v8f S = __builtin_amdgcn_wmma_f32_16x16x4_f32(false, A, false, Bm, (short)0, C,
                                                false, false);
  #pragma unroll
  for (int r = 0; r < 8; ++r) {
    int m = r + halfOff;
    int gi = i0 + m, gj = j0 + n;
    float rx1 = sh[m*4+0], ry1 = sh[m*4+1], rx2 = sh[m*4+2], ry2 = sh[m*4+3];
    float xx1 = fmaxf(rx1, cx1), yy1 = fmaxf(ry1, cy1);
    float xx2 = fminf(rx2, cx2), yy2 = fminf(ry2, cy2);
    float inter = fmaxf(xx2 - xx1 + 1.0f, 0.0f) * fmaxf(yy2 - yy1 + 1.0f, 0.0f);
    float iou = inter / (S[r] - inter);
    bool pred = (iou > NMS_TH) && (gj > gi);
    u32 bal = __builtin_amdgcn_ballot_w32(pred);
    if (lane == 0)      mask16[(size_t)(i0 + r)     * COLS16 + jt] = (u16)(bal & 0xFFFFu);
    else if (lane == 1) mask16[(size_t)(i0 + r + 8) * COLS16 + jt] = (u16)(bal >> 16);
  }
}

// ---------------- single-wave greedy scan (per-wave in-order LDS) ------------
__global__ __launch_bounds__(32) void k_scan(const u16* __restrict__ maskBase,
                                             u32* __restrict__ keep) {
  int z = blockIdx.x;
  const u32* mask = (const u32*)(maskBase + (size_t)z * K_PRE * COLS16);
  __shared__ u32 sup[192];
  volatile u32* vsup = sup;
  int lane = threadIdx.x;
  #pragma unroll
  for (int s = 0; s < 6; ++s) sup[lane + 32 * s] = 0;
  asm volatile("" ::: "memory");
  for (int i = 0; i < K_PRE; ++i) {
    u32 sw = vsup[i >> 5];                   // all lanes same word -> uniform branch
    if (!((sw >> (i & 31)) & 1u)) {
      const u32* row = mask + (size_t)i * WORDS;
      int wstart = i >> 5;
      bool oddHalf = (i >> 4) & 1;
      #pragma unroll
      for (int s = 0; s < 6; ++s) {
        int w = lane + 32 * s;
        if (w >= wstart && w < WORDS) {
          u32 m = row[w];
          if (w == wstart && oddHalf) m &= 0xFFFF0000u;  // lower-tri u16 never written
          if (w == WORDS - 1)         m &= 0x0000FFFFu;  // pad column never written
          if (m) atomicOr(&sup[w], m);
        }
      }
    }
    asm volatile("" ::: "memory");
  }
  #pragma unroll
  for (int s = 0; s < 6; ++s) {
    int w = lane + 32 * s;
    if (w < WORDS) {
      u32 kv = ~vsup[w];
      if (w == WORDS - 1) kv &= 0x0000FFFFu;
      keep[(size_t)z * 192 + w] = kv;
    }
  }
}

// ---------------- combine keeps, rank, emit [b, x1,y1,x2,y2] -----------------
__global__ void k_output(const u32* __restrict__ keep,
                         const float* __restrict__ boxesL, const float* __restrict__ boxesR,
                         float* __restrict__ outL, float* __restrict__ outR) {
  int b = blockIdx.x, t = threadIdx.x;
  __shared__ u32 comb[WORDS];
  __shared__ u32 pref[WORDS];
  if (t < WORDS)
    comb[t] = keep[(size_t)(2*b) * 192 + t] & keep[(size_t)(2*b + 1) * 192 + t];
  __syncthreads();
  if (t == 0) {
    u32 run = 0;
    for (int w = 0; w < WORDS; ++w) { pref[w] = run; run += __popc(comb[w]); }
  }
  __syncthreads();
  for (int r = t; r < K_POST; r += blockDim.x) {
    float* ol = outL + ((size_t)b * K_POST + r) * 5;
    float* og = outR + ((size_t)b * K_POST + r) * 5;
    ol[0] = (float)b; ol[1]=0.f; ol[2]=0.f; ol[3]=0.f; ol[4]=0.f;
    og[0] = (float)b; og[1]=0.f; og[2]=0.f; og[3]=0.f; og[4]=0.f;
  }
  __syncthreads();
  for (int i = t; i < K_PRE; i += blockDim.x) {
    u32 w = i >> 5, c = comb[w];
    if ((c >> (i & 31)) & 1u) {
      u32 rank = pref[w] + __popc(c & ((1u << (i & 31)) - 1u));
      if (rank < K_POST) {
        const float* bl = boxesL + ((size_t)b * K_PRE + i) * 4;
        const float* br = boxesR + ((size_t)b * K_PRE + i) * 4;
        float* ol = outL + ((size_t)b * K_POST + rank) * 5;
        float* og = outR + ((size_t)b * K_POST + rank) * 5;
        ol[1]=bl[0]; ol[2]=bl[1]; ol[3]=bl[2]; ol[4]=bl[3];
        og[1]=br[0]; og[2]=br[1]; og[3]=br[2]; og[4]=br[3];
      }
    }
  }
}

extern "C" void kernel_launch(void* const* d_in, const int* in_sizes, int n_in,
                              void* d_out, int out_size, void* d_ws, size_t ws_size,
                              hipStream_t stream) {
  const float* scores  = (const float*)d_in[0];   // (B, N, 2)
  const float* deltas  = (const float*)d_in[1];   // (B, N, 6)
  const float* im_info = (const float*)d_in[2];   // (B, 3)
  const float* anchors = (const float*)d_in[3];   // (N, 4)
  int B = in_sizes[2] / 3;
  int N = in_sizes[3] / 4;

  char* p = (char*)d_ws;
  auto alloc = [&](size_t bytes) -> void* {
    void* r = (void*)p;
    p += (bytes + 255) & ~(size_t)255;
    return r;
  };
  u32* ukeys   = (u32*)alloc((size_t)B * N * 4);
  u32* hist    = (u32*)alloc((size_t)B * 256 * 4);
  u32* state   = (u32*)alloc((size_t)B * 2 * 4);
  u32* cnt     = (u32*)alloc((size_t)B * 4);
  u64* comp    = (u64*)alloc((size_t)B * 8192 * 8);
  u32* order   = (u32*)alloc((size_t)B * K_PRE * 4);
  float* boxesL = (float*)alloc((size_t)B * K_PRE * 4 * 4);
  float* areasL = (float*)alloc((size_t)B * K_PRE * 4);
  float* boxesR = (float*)alloc((size_t)B * K_PRE * 4 * 4);
  float* areasR = (float*)alloc((size_t)B * K_PRE * 4);
  u16* mask    = (u16*)alloc((size_t)2 * B * K_PRE * COLS16 * 2);
  u32* keep    = (u32*)alloc((size_t)2 * B * 192 * 4);
  (void)ws_size; (void)n_in; (void)out_size;

  k_init<<<1, 256, 0, stream>>>(hist, state, cnt, B);
  int totalBN = B * N;
  k_keys<<<(totalBN + 255) / 256, 256, 0, stream>>>(scores, ukeys, totalBN);
  dim3 gh((N + 255) / 256, B);
  for (int lvl = 0; lvl < 4; ++lvl) {
    k_hist<<<gh, 256, 0, stream>>>(ukeys, hist, state, lvl, N);
    k_find<<<B, 256, 0, stream>>>(hist, state);
  }
  k_compact<<<gh, 256, 0, stream>>>(ukeys, state, cnt, comp, N);
  k_sort<<<B, 1024, 0, stream>>>(comp, cnt, order);
  int totalBK = B * K_PRE;
  k_gather<<<(totalBK + 255) / 256, 256, 0, stream>>>(order, anchors, deltas, im_info,
                                                      boxesL, areasL, boxesR, areasR,
                                                      N, totalBK);
  k_mask<<<dim3(TILES, TILES, 2 * B), 32, 0, stream>>>(boxesL, areasL, boxesR, areasR, mask);
  k_scan<<<2 * B, 32, 0, stream>>>(mask, keep);
  float* outL = (float*)d_out;
  float* outR = outL + (size_t)B * K_POST * 5;
  k_output<<<B, 256, 0, stream>>>(keep, boxesL, boxesR, outL, outR);
}